// SCDecoder_86157043958172
// MI455X (gfx1250) — compile-verified
//
#include <hip/hip_runtime.h>

#define POLAR_N 256

// fast softplus: log1p(exp(-x)) for x >= 0, via v_exp_f32 / v_log_f32.
__device__ __forceinline__ float sp_neg(float x) {
    return __logf(1.0f + __expf(-x));
}

// boxplus(a,b) = logaddexp(a,b) - logaddexp(a+b,0)
//             = max(a,b) - max(a+b,0) + log1p(e^-|a-b|) - log1p(e^-|a+b|)
__device__ __forceinline__ float bxp(float a, float b) {
    float d = fabsf(a - b);
    float s = a + b;
    return fmaxf(a, b) - fmaxf(s, 0.0f) + sp_neg(d) - sp_neg(fabsf(s));
}

// g-function: b + (bit ? -a : a) via sign-bit xor
__device__ __forceinline__ float gop(float a, float b, unsigned bit) {
    return b + __int_as_float(__float_as_int(a) ^ (bit << 31));
}

__device__ __forceinline__ unsigned part1by1(unsigned x) {
    x &= 0x0000FFFFu;
    x = (x | (x << 8)) & 0x00FF00FFu;
    x = (x | (x << 4)) & 0x0F0F0F0Fu;
    x = (x | (x << 2)) & 0x33333333u;
    x = (x | (x << 1)) & 0x55555555u;
    return x;
}

// interleave: out bit 2i = e[i], bit 2i+1 = o[i]
__device__ __forceinline__ void ilv(unsigned e, unsigned o, unsigned &lo, unsigned &hi) {
    lo = part1by1(e & 0xFFFFu) | (part1by1(o & 0xFFFFu) << 1);
    hi = part1by1(e >> 16)     | (part1by1(o >> 16)     << 1);
}

// packed-bit base word per depth: d0:0(8w) d1:8(4w) d2:12(2w) d3..8:14..19(1w)
__device__ __forceinline__ int cbase(int d) {
    return (d >= 3) ? (d + 11) : ((d == 2) ? 12 : ((d == 1) ? 8 : 0));
}

// per-chain LDS LLR stack offsets, depths 2..4 only (64,32,16 floats); 5..7 live in regs
#define LOFF(d) (128 - (512 >> (d)))
#define LSTRIDE 114   // -14*lane mod 64 -> 32 distinct even bank residues, conflict free

__global__ __launch_bounds__(32) void polar_sc_kernel(
    const float* __restrict__ ex, const float* __restrict__ ey,
    const float* __restrict__ f,  const float* __restrict__ r,
    float* __restrict__ out)
{
    __shared__ float    sL[32 * LSTRIDE]; // 14592 B : LLR stacks d2..d4 (one chain per lane)
    __shared__ unsigned sC[32 * 21];      //  2688 B : packed decoded bits (stride 21, odd)
    __shared__ float    sF[16 * 260];     // 16640 B : staged f rows (stride 260, no conflicts)
    __shared__ float    sR[16 * 260];     // 16640 B : staged r rows

    const int lane = threadIdx.x;
    const int sub  = lane & 1;            // 0 = ex chain, 1 = ey chain
    const int rid  = lane >> 1;           // row within block (16 rows/block)
    const size_t row = (size_t)blockIdx.x * 16 + rid;

    const float* __restrict__ crow = (sub ? ey : ex) + row * POLAR_N;
    const float4* __restrict__ c4  = (const float4*)crow;

    float*    Lc = sL + lane * LSTRIDE;
    unsigned* Cc = sC + lane * 21;

    // CDNA5 global_prefetch_b8: warm this chain's row into GL2
#pragma unroll
    for (int j = 0; j < 8; ++j)
        __builtin_prefetch(crow + j * 32, 0, 3);

    // stage f/r for the block's 16 rows, coalesced b128 loads
    {
        const float4* __restrict__ f4 = (const float4*)(f + (size_t)blockIdx.x * 16 * POLAR_N);
        const float4* __restrict__ r4 = (const float4*)(r + (size_t)blockIdx.x * 16 * POLAR_N);
        for (int i4 = lane; i4 < 1024; i4 += 32) {
            int rr_ = i4 >> 6, j4 = i4 & 63;
            ((float4*)(sF + rr_ * 260))[j4] = f4[i4];
            ((float4*)(sR + rr_ * 260))[j4] = r4[i4];
        }
        __syncthreads();
    }

    // register-resident deep levels (persist across phi iterations)
    float l5_0 = 0.f, l5_1 = 0.f, l5_2 = 0.f, l5_3 = 0.f,
          l5_4 = 0.f, l5_5 = 0.f, l5_6 = 0.f, l5_7 = 0.f;
    float l6_0 = 0.f, l6_1 = 0.f, l6_2 = 0.f, l6_3 = 0.f;
    float l7_0 = 0.f, l7_1 = 0.f;
    float l8 = 0.f;

    for (int phi = 0; phi < POLAR_N; ++phi) {
        int dstart; bool startg;
        if ((phi & 63) == 0) {
            // fused recompute of depth-1 (not stored) and depth-2 from global inputs
            bool l1g = (phi >= 128);
            bool l2g = (phi & 64) != 0;
            unsigned c1w0 = 0, c1w1 = 0, c1w2 = 0, c1w3 = 0, c2w0 = 0, c2w1 = 0;
            if (l1g) { c1w0 = Cc[8]; c1w1 = Cc[9]; c1w2 = Cc[10]; c1w3 = Cc[11]; }
            if (l2g) { c2w0 = Cc[12]; c2w1 = Cc[13]; }
            for (int i = 0; i < 64; ++i) {
                float4 v = c4[i];
                float l10, l11;
                if (l1g) {
                    unsigned w = (i < 16) ? c1w0 : ((i < 32) ? c1w1 : ((i < 48) ? c1w2 : c1w3));
                    unsigned sh = ((unsigned)(2 * i)) & 31u;
                    l10 = gop(v.x, v.y, (w >> sh) & 1u);
                    l11 = gop(v.z, v.w, (w >> (sh + 1)) & 1u);
                } else {
                    l10 = bxp(v.x, v.y);
                    l11 = bxp(v.z, v.w);
                }
                if (l2g) {
                    unsigned w2 = (i < 32) ? c2w0 : c2w1;
                    Lc[LOFF(2) + i] = gop(l10, l11, (w2 >> (i & 31)) & 1u);
                } else {
                    Lc[LOFF(2) + i] = bxp(l10, l11);
                }
            }
            dstart = 3; startg = false;
        } else {
            int t = __builtin_ctz(phi);   // t <= 5 here
            dstart = 8 - t;
            startg = true;
        }

        // generic LDS -> LDS levels: dc = 3..4 (h = 32, 16)
        for (int dc = dstart; dc <= 4; ++dc) {
            bool gg = startg && (dc == dstart);
            unsigned cw = gg ? Cc[cbase(dc)] : 0u;
            int h = POLAR_N >> dc;
            float* __restrict__ dst = Lc + LOFF(dc);
            const float* __restrict__ src = Lc + LOFF(dc - 1);
            for (int i = 0; i < h; i += 2) {
                float2 p0 = *(const float2*)(src + 2 * i);
                float2 p1 = *(const float2*)(src + 2 * i + 2);
                if (gg) {
                    dst[i]     = gop(p0.x, p0.y, (cw >> i) & 1u);
                    dst[i + 1] = gop(p1.x, p1.y, (cw >> (i + 1)) & 1u);
                } else {
                    dst[i]     = bxp(p0.x, p0.y);
                    dst[i + 1] = bxp(p1.x, p1.y);
                }
            }
        }

        // level 5: LDS d4 -> registers (8 outputs)
        if (dstart <= 5) {
            bool gg = startg && (dstart == 5);
            unsigned cw = gg ? Cc[16] : 0u;
            const float* s4 = Lc + LOFF(4);
            float2 q0 = *(const float2*)(s4 + 0);
            float2 q1 = *(const float2*)(s4 + 2);
            float2 q2 = *(const float2*)(s4 + 4);
            float2 q3 = *(const float2*)(s4 + 6);
            float2 q4 = *(const float2*)(s4 + 8);
            float2 q5 = *(const float2*)(s4 + 10);
            float2 q6 = *(const float2*)(s4 + 12);
            float2 q7 = *(const float2*)(s4 + 14);
            if (gg) {
                l5_0 = gop(q0.x, q0.y,  cw       & 1u);
                l5_1 = gop(q1.x, q1.y, (cw >> 1) & 1u);
                l5_2 = gop(q2.x, q2.y, (cw >> 2) & 1u);
                l5_3 = gop(q3.x, q3.y, (cw >> 3) & 1u);
                l5_4 = gop(q4.x, q4.y, (cw >> 4) & 1u);
                l5_5 = gop(q5.x, q5.y, (cw >> 5) & 1u);
                l5_6 = gop(q6.x, q6.y, (cw >> 6) & 1u);
                l5_7 = gop(q7.x, q7.y, (cw >> 7) & 1u);
            } else {
                l5_0 = bxp(q0.x, q0.y); l5_1 = bxp(q1.x, q1.y);
                l5_2 = bxp(q2.x, q2.y); l5_3 = bxp(q3.x, q3.y);
                l5_4 = bxp(q4.x, q4.y); l5_5 = bxp(q5.x, q5.y);
                l5_6 = bxp(q6.x, q6.y); l5_7 = bxp(q7.x, q7.y);
            }
        }
        // level 6: registers -> registers (4 outputs)
        if (dstart <= 6) {
            bool gg = startg && (dstart == 6);
            unsigned cw = gg ? Cc[17] : 0u;
            if (gg) {
                l6_0 = gop(l5_0, l5_1,  cw       & 1u);
                l6_1 = gop(l5_2, l5_3, (cw >> 1) & 1u);
                l6_2 = gop(l5_4, l5_5, (cw >> 2) & 1u);
                l6_3 = gop(l5_6, l5_7, (cw >> 3) & 1u);
            } else {
                l6_0 = bxp(l5_0, l5_1); l6_1 = bxp(l5_2, l5_3);
                l6_2 = bxp(l5_4, l5_5); l6_3 = bxp(l5_6, l5_7);
            }
        }
        // level 7: registers -> registers (2 outputs)
        if (dstart <= 7) {
            bool gg = startg && (dstart == 7);
            unsigned cw = gg ? Cc[18] : 0u;
            if (gg) {
                l7_0 = gop(l6_0, l6_1,  cw       & 1u);
                l7_1 = gop(l6_2, l6_3, (cw >> 1) & 1u);
            } else {
                l7_0 = bxp(l6_0, l6_1); l7_1 = bxp(l6_2, l6_3);
            }
        }
        // level 8 (leaf LLR)
        {
            if (startg && dstart == 8) l8 = gop(l7_0, l7_1, Cc[19] & 1u);
            else                       l8 = bxp(l7_0, l7_1);
        }

        // ---- leaf: exchange chains within the lane pair ----
        float mine  = l8;
        float other = __shfl_xor(mine, 1, 32);
        float llr_u  = sub ? other : mine;
        float llr_uy = sub ? mine  : other;

        float fv = sF[rid * 260 + phi];
        float rr = sR[rid * 260 + phi];
        float pu = 1.0f / (1.0f + expf(-llr_u));     // accurate exp for the sampler
        float frozen = (rr >= pu) ? 0.0f : 1.0f;
        bool  isf = (fv == 0.0f) || (fv == 1.0f);
        float xh = (llr_uy > 0.0f) ? 1.0f : 0.0f;
        float xb = isf ? frozen : xh;

        // split the 3 output stores across the lane pair (both lanes have all values)
        {
            float* o = out + (row * POLAR_N + (size_t)phi) * 4;
            if (!sub) *(float2*)(o + 2) = make_float2(llr_u, llr_uy);
            else      o[0] = xb;                         // u
        }

        // ---- propagate decoded bit: x = interleave(x1 ^ x2, x2) while right child ----
        unsigned v0 = (unsigned)(xb != 0.0f), v1 = 0, v2 = 0, v3 = 0,
                 v4 = 0, v5 = 0, v6 = 0, v7 = 0;
        int size = 1, d = 8, ph = phi;
        while (ph & 1) {
            int base = cbase(d);
            if (size < 32) {
                unsigned A = Cc[base];
                unsigned mask = (1u << size) - 1u;
                unsigned lo, hi;
                ilv((A ^ v0) & mask, v0 & mask, lo, hi);
                v0 = lo; (void)hi;
            } else if (size == 32) {
                unsigned A = Cc[base];
                unsigned t0, t1; ilv(A ^ v0, v0, t0, t1);
                v0 = t0; v1 = t1;
            } else if (size == 64) {
                unsigned A0 = Cc[base], A1 = Cc[base + 1];
                unsigned t0, t1, t2, t3;
                ilv(A0 ^ v0, v0, t0, t1); ilv(A1 ^ v1, v1, t2, t3);
                v0 = t0; v1 = t1; v2 = t2; v3 = t3;
            } else { // 128
                unsigned A0 = Cc[base], A1 = Cc[base + 1];
                unsigned A2 = Cc[base + 2], A3 = Cc[base + 3];
                unsigned t0, t1, t2, t3, t4, t5, t6, t7;
                ilv(A0 ^ v0, v0, t0, t1); ilv(A1 ^ v1, v1, t2, t3);
                ilv(A2 ^ v2, v2, t4, t5); ilv(A3 ^ v3, v3, t6, t7);
                v0 = t0; v1 = t1; v2 = t2; v3 = t3; v4 = t4; v5 = t5; v6 = t6; v7 = t7;
            }
            size <<= 1; d -= 1; ph >>= 1;
        }
        {
            int base = cbase(d);
            Cc[base] = v0;
            if (size > 32)  { Cc[base + 1] = v1; }
            if (size > 64)  { Cc[base + 2] = v2; Cc[base + 3] = v3; }
            if (size > 128) { Cc[base + 4] = v4; Cc[base + 5] = v5;
                              Cc[base + 6] = v6; Cc[base + 7] = v7; }
        }
    }

    // ---- write x (channel order) from root bit array C[0]; split across the pair ----
#pragma unroll
    for (int k = 0; k < 4; ++k) {
        unsigned wk = Cc[(sub ? 4 : 0) + k];
        int ib = (sub ? 128 : 0) + k * 32;
        for (int b = 0; b < 32; ++b) {
            out[(row * POLAR_N + (size_t)(ib + b)) * 4 + 1] =
                (float)((wk >> b) & 1u);
        }
    }
}

extern "C" void kernel_launch(void* const* d_in, const int* in_sizes, int n_in,
                              void* d_out, int out_size, void* d_ws, size_t ws_size,
                              hipStream_t stream) {
    (void)n_in; (void)out_size; (void)d_ws; (void)ws_size;
    const float* ex = (const float*)d_in[0];
    const float* ey = (const float*)d_in[1];
    const float* f  = (const float*)d_in[2];
    const float* r  = (const float*)d_in[3];
    float* out = (float*)d_out;
    int B = in_sizes[0] / POLAR_N;
    int blocks = (B + 15) / 16;          // 2 lanes (ex/ey chains) per row, 16 rows/block
    hipLaunchKernelGGL(polar_sc_kernel, dim3(blocks), dim3(32), 0, stream,
                       ex, ey, f, r, out);
}